// LabelSmoothingLoss_16406775071064
// MI455X (gfx1250) — compile-verified
//
#include <hip/hip_runtime.h>
#include <stdint.h>

#define N_ROWS   4096
#define C_DIM    50257
#define C4_FULL  12564          // 50256 / 4 full float4s; element 50256 is a scalar tail
#define TILE4    1536           // float4s per LDS tile (24 KB)
#define FULL_TILES 8            // 8 * 1536 = 12288 float4s via async-LDS path
#define TAIL4    (C4_FULL - FULL_TILES * TILE4)   // 276 float4s via plain loads
#define THREADS  256
#define PER_THREAD (TILE4 / THREADS)              // 6 async b128 per thread per tile
#define SMOOTH   0.1f

// CDNA5 async global->LDS copy, 128-bit, non-temporal (single-use stream).
__device__ __forceinline__ void async_ld_f4(uint32_t lds_off, uint64_t gaddr) {
  asm volatile("global_load_async_to_lds_b128 %0, %1, off th:TH_LOAD_NT"
               :: "v"(lds_off), "v"(gaddr) : "memory");
}

__global__ __launch_bounds__(THREADS)
void LabelSmoothingLoss_row_kernel(const float* __restrict__ x,
                                   const int* __restrict__ tgt,
                                   float* __restrict__ row_loss) {
  __shared__ float4 buf[2][TILE4];          // 48 KB double buffer
  __shared__ float wm[THREADS / 32], wsum[THREADS / 32], wt[THREADS / 32];

  const int row = blockIdx.x;
  const int tid = threadIdx.x;
  const float* rowp = x + (size_t)row * C_DIM;
  const uint64_t gbase = (uint64_t)(uintptr_t)rowp;
  uint32_t lbase[2];
  lbase[0] = (uint32_t)(uintptr_t)(&buf[0][0]);  // generic addr low 32 bits = LDS offset
  lbase[1] = (uint32_t)(uintptr_t)(&buf[1][0]);

  // Prime the pipeline: issue tile 0 (6 async instructions per wave).
  {
    uint32_t l = lbase[0] + (uint32_t)tid * 16u;
    uint64_t g = gbase + (uint64_t)tid * 16u;
#pragma unroll
    for (int k = 0; k < PER_THREAD; ++k)
      async_ld_f4(l + (uint32_t)k * THREADS * 16u, g + (uint64_t)k * THREADS * 16u);
  }

  float m = -3.0e38f, s = 0.0f, tsum = 0.0f;

  for (int tIdx = 0; tIdx < FULL_TILES; ++tIdx) {
    if (tIdx + 1 < FULL_TILES) {
      // Issue next tile into the other buffer (free: everyone passed last barrier).
      const int nb = (tIdx + 1) & 1;
      uint32_t l = lbase[nb] + (uint32_t)tid * 16u;
      uint64_t g = gbase + ((uint64_t)(tIdx + 1) * TILE4 + (uint64_t)tid) * 16u;
#pragma unroll
      for (int k = 0; k < PER_THREAD; ++k)
        async_ld_f4(l + (uint32_t)k * THREADS * 16u, g + (uint64_t)k * THREADS * 16u);
      // 6 newest async ops per wave belong to tile t+1; in-order completion =>
      // <=6 outstanding means tile t has fully landed in LDS for this wave.
      asm volatile("s_wait_asynccnt 0x6" ::: "memory");
    } else {
      asm volatile("s_wait_asynccnt 0x0" ::: "memory");
    }
    __syncthreads();

    // Consume tile t from LDS into registers; one exp per element.
    const float4* lb = &buf[tIdx & 1][0];
    float v[PER_THREAD * 4];
#pragma unroll
    for (int k = 0; k < PER_THREAD; ++k) {
      float4 f = lb[tid + k * THREADS];
      v[4 * k + 0] = f.x; v[4 * k + 1] = f.y;
      v[4 * k + 2] = f.z; v[4 * k + 3] = f.w;
    }
    float mt = v[0];
#pragma unroll
    for (int k = 1; k < PER_THREAD * 4; ++k) mt = fmaxf(mt, v[k]);
    float st = 0.0f;
#pragma unroll
    for (int k = 0; k < PER_THREAD * 4; ++k) {
      st += __expf(v[k] - mt);
      tsum += v[k];
    }
    const float M = fmaxf(m, mt);
    s = s * __expf(m - M) + st * __expf(mt - M);
    m = M;
    __syncthreads();  // buffer t%2 free for reuse at t+2
  }

  // Row tail: 276 float4s through plain vector loads (online 2-exp update).
  const float4* rowp4 = (const float4*)rowp;
  for (int i = tid; i < TAIL4; i += THREADS) {
    float4 f = rowp4[FULL_TILES * TILE4 + i];
    float vv[4] = {f.x, f.y, f.z, f.w};
#pragma unroll
    for (int k = 0; k < 4; ++k) {
      const float M = fmaxf(m, vv[k]);
      s = s * __expf(m - M) + __expf(vv[k] - M);
      m = M; tsum += vv[k];
    }
  }
  // Final scalar element (index C-1 = 50256).
  if (tid == 0) {
    const float vv = rowp[C_DIM - 1];
    const float M = fmaxf(m, vv);
    s = s * __expf(m - M) + __expf(vv - M);
    m = M; tsum += vv;
  }

  // Wave32 reduction of (m, s, tsum).
#pragma unroll
  for (int off = 16; off > 0; off >>= 1) {
    const float om = __shfl_xor(m, off, 32);
    const float os = __shfl_xor(s, off, 32);
    const float ot = __shfl_xor(tsum, off, 32);
    const float M = fmaxf(m, om);
    s = s * __expf(m - M) + os * __expf(om - M);
    m = M; tsum += ot;
  }
  const int wave = tid >> 5;
  if ((tid & 31) == 0) { wm[wave] = m; wsum[wave] = s; wt[wave] = tsum; }
  __syncthreads();

  if (tid == 0) {
    float M = wm[0], S = wsum[0], T = wt[0];
#pragma unroll
    for (int w = 1; w < THREADS / 32; ++w) {
      const float M2 = fmaxf(M, wm[w]);
      S = S * __expf(M - M2) + wsum[w] * __expf(wm[w] - M2);
      M = M2; T += wt[w];
    }
    const float lse = M + logf(S);
    const float off = SMOOTH / (float)(C_DIM - 1);
    const float wgt = 1.0f - SMOOTH - off;
    const float xt = rowp[tgt[row]];
    const float tgt_lp = xt - lse;
    const float sum_lp = T - (float)C_DIM * lse;
    row_loss[row] = -(wgt * tgt_lp + off * sum_lp);
  }
}

__global__ __launch_bounds__(THREADS)
void LabelSmoothingLoss_mean_kernel(const float* __restrict__ row_loss,
                                    float* __restrict__ out) {
  __shared__ float partial[THREADS / 32];
  const int tid = threadIdx.x;
  float a = 0.0f;
  for (int i = tid; i < N_ROWS; i += THREADS) a += row_loss[i];  // fixed order
#pragma unroll
  for (int off = 16; off > 0; off >>= 1) a += __shfl_xor(a, off, 32);
  if ((tid & 31) == 0) partial[tid >> 5] = a;
  __syncthreads();
  if (tid == 0) {
    float sum = 0.0f;
#pragma unroll
    for (int w = 0; w < THREADS / 32; ++w) sum += partial[w];
    out[0] = sum * (1.0f / (float)N_ROWS);
  }
}

extern "C" void kernel_launch(void* const* d_in, const int* in_sizes, int n_in,
                              void* d_out, int out_size, void* d_ws, size_t ws_size,
                              hipStream_t stream) {
  (void)in_sizes; (void)n_in; (void)out_size; (void)ws_size;
  const float* x   = (const float*)d_in[0];
  const int*   tgt = (const int*)d_in[1];
  float* row_loss  = (float*)d_ws;           // 4096 floats of scratch
  float* out       = (float*)d_out;

  LabelSmoothingLoss_row_kernel<<<N_ROWS, THREADS, 0, stream>>>(x, tgt, row_loss);
  LabelSmoothingLoss_mean_kernel<<<1, THREADS, 0, stream>>>(row_loss, out);
}